// CurvDist_24790551233442
// MI455X (gfx1250) — compile-verified
//
#include <hip/hip_runtime.h>
#include <math.h>

typedef __attribute__((ext_vector_type(2))) float v2f;
typedef __attribute__((ext_vector_type(8))) float v8f;

#define NPTS  4096
#define NTILES (NPTS / 16)

// Branchless sorted-top3 insert. Invariant: (v0,i0) <= (v1,i1) <= (v2,i2).
// New sorted values via min/med3 (1 v_min_f32 + 2 v_med3_f32); indices via
// 3 compares + 5 cndmask. On exact value ties the incumbent wins, which —
// since candidates arrive in increasing index order — matches top_k's
// first-occurrence tie-breaking.
__device__ __forceinline__ void top3_insert(float v, int idx,
                                            float& v0, float& v1, float& v2,
                                            int& i0, int& i1, int& i2) {
  bool c0 = v < v0;
  bool c1 = v < v1;
  bool c2 = v < v2;
  float nv0 = fminf(v, v0);
  float nv1 = __builtin_amdgcn_fmed3f(v0, v1, v);
  float nv2 = __builtin_amdgcn_fmed3f(v1, v2, v);
  int ni0 = c0 ? idx : i0;
  int ni1 = c0 ? i0 : (c1 ? idx : i1);
  int ni2 = c1 ? i1 : (c2 ? idx : i2);
  v0 = nv0; v1 = nv1; v2 = nv2;
  i0 = ni0; i1 = ni1; i2 = ni2;
}

// Scan all NPTS database points P (layout [3][NPTS]) against the 16 queries
// encoded in the B operand bq; returns per-query global top-3 of the shifted
// distance s(q,p) = |p|^2 - 2 p.q  (ordering-equivalent to squared distance).
//
// WMMA D = A(16x4) x B(4x16):
//   A lane L<16:  {p.x, p.y}    lane L+16: {p.z, |p|^2}     (K pairs {0,1}/{2,3})
//   B lane L<16:  {-2q.x,-2q.y} lane L+16: {-2q.z, 1}
//   C/D: lane L<16 holds rows M=0..7 of column L in d[0..7];
//        lane L+16 holds rows M=8..15 of column L.
__device__ __forceinline__ void scan_top3(const float* __restrict__ P,
                                          v2f bq, int lane,
                                          float& v0, float& v1, float& v2,
                                          int& i0, int& i1, int& i2) {
  v0 = v1 = v2 = 3.0e38f;
  i0 = i1 = i2 = 0x7fffffff;
  const int  l15 = lane & 15;
  const bool hi  = lane >= 16;

  for (int t = 0; t < NTILES; ++t) {
    int p = t * 16 + l15;
    float px = P[p];
    float py = P[NPTS + p];
    float pz = P[2 * NPTS + p];
    v2f a;
    a.x = hi ? pz : px;                                   // branchless selects
    a.y = hi ? (px * px + py * py + pz * pz) : py;
    v8f c = {};
    v8f d = __builtin_amdgcn_wmma_f32_16x16x4_f32(
        /*neg_a=*/false, a, /*neg_b=*/false, bq,
        /*c_mod=*/(short)0, c, /*reuse_a=*/false, /*reuse_b=*/false);
    int base = t * 16 + (hi ? 8 : 0);
#pragma unroll
    for (int r = 0; r < 8; ++r) {
      top3_insert(d[r], base + r, v0, v1, v2, i0, i1, i2);
    }
  }

  // Merge the two half-row partial top-3 lists held by lanes (L, L^16).
  float ov0 = __shfl_xor(v0, 16, 32);
  float ov1 = __shfl_xor(v1, 16, 32);
  float ov2 = __shfl_xor(v2, 16, 32);
  int   oi0 = __shfl_xor(i0, 16, 32);
  int   oi1 = __shfl_xor(i1, 16, 32);
  int   oi2 = __shfl_xor(i2, 16, 32);
  top3_insert(ov0, oi0, v0, v1, v2, i0, i1, i2);
  top3_insert(ov1, oi1, v0, v1, v2, i0, i1, i2);
  top3_insert(ov2, oi2, v0, v1, v2, i0, i1, i2);
}

__device__ __forceinline__ v2f make_query_operand(const float* __restrict__ Q,
                                                  int q, int lane,
                                                  float& qx, float& qy, float& qz) {
  qx = Q[q];
  qy = Q[NPTS + q];
  qz = Q[2 * NPTS + q];
  bool hi = lane >= 16;
  v2f bq;
  bq.x = hi ? (-2.0f * qz) : (-2.0f * qx);
  bq.y = hi ? 1.0f : (-2.0f * qy);
  return bq;
}

__device__ __forceinline__ float kappa_from_neighbors(const float* __restrict__ P,
                                                      int n1, int n2,
                                                      float qx, float qy, float qz,
                                                      float nx, float ny, float nz) {
  float acc = 0.0f;
#pragma unroll
  for (int j = 0; j < 2; ++j) {
    int p = (j == 0) ? n1 : n2;
    float dx = P[p] - qx;
    float dy = P[NPTS + p] - qy;
    float dz = P[2 * NPTS + p] - qz;
    float len = sqrtf(dx * dx + dy * dy + dz * dz);
    float inv = 1.0f / fmaxf(len, 1e-12f);
    acc += fabsf((dx * nx + dy * ny + dz * nz) * inv);
  }
  return 0.5f * acc;
}

// Kernel 1: kappa of the original cloud with its own normals.
__global__ void __launch_bounds__(32)
ori_kappa_kernel(const float* __restrict__ ori, const float* __restrict__ nrm,
                 float* __restrict__ kappa_out) {
  int wg   = blockIdx.x;
  int b    = wg / NTILES;
  int ta   = wg % NTILES;
  int lane = threadIdx.x;
  const float* P  = ori + (size_t)b * 3 * NPTS;
  const float* NR = nrm + (size_t)b * 3 * NPTS;
  int l15 = lane & 15;
  int q   = ta * 16 + l15;

  float qx, qy, qz;
  v2f bq = make_query_operand(P, q, lane, qx, qy, qz);

  float v0, v1, v2;
  int i0, i1, i2;
  scan_top3(P, bq, lane, v0, v1, v2, i0, i1, i2);   // i0 == self

  if (lane < 16) {
    float nx = NR[q], ny = NR[NPTS + q], nz = NR[2 * NPTS + q];
    kappa_out[(size_t)b * NPTS + q] =
        kappa_from_neighbors(P, i1, i2, qx, qy, qz, nx, ny, nz);
  }
}

// Kernel 2: adv->ori 1-NN, adv self-kNN, adv kappa with gathered normals,
// per-point squared error vs 1-NN ori kappa.
__global__ void __launch_bounds__(32)
adv_err_kernel(const float* __restrict__ ori, const float* __restrict__ adv,
               const float* __restrict__ nrm, const float* __restrict__ ori_kappa,
               float* __restrict__ sqerr_out) {
  int wg   = blockIdx.x;
  int b    = wg / NTILES;
  int ta   = wg % NTILES;
  int lane = threadIdx.x;
  const float* PO = ori + (size_t)b * 3 * NPTS;
  const float* PA = adv + (size_t)b * 3 * NPTS;
  const float* NR = nrm + (size_t)b * 3 * NPTS;
  int l15 = lane & 15;
  int q   = ta * 16 + l15;

  float qx, qy, qz;
  v2f bq = make_query_operand(PA, q, lane, qx, qy, qz);

  // Pass A: nearest ori point for each adv query (argmin == i0 of top-3).
  float v0, v1, v2;
  int i0, i1, i2;
  scan_top3(PO, bq, lane, v0, v1, v2, i0, i1, i2);
  int intra = i0;   // all 32 lanes hold the merged result — no divergence

  // Pass B: self-kNN of the adv cloud (i0 == self; neighbors i1, i2).
  scan_top3(PA, bq, lane, v0, v1, v2, i0, i1, i2);

  if (lane < 16) {
    float nx = NR[intra], ny = NR[NPTS + intra], nz = NR[2 * NPTS + intra];
    float ka = kappa_from_neighbors(PA, i1, i2, qx, qy, qz, nx, ny, nz);
    float diff = ka - ori_kappa[(size_t)b * NPTS + intra];
    sqerr_out[(size_t)b * NPTS + q] = diff * diff;
  }
}

// Kernel 3: mean over all B*N squared errors.
__global__ void __launch_bounds__(256)
reduce_kernel(const float* __restrict__ sqerr, float* __restrict__ out, int n) {
  __shared__ float sm[256];
  float s = 0.0f;
  for (int i = threadIdx.x; i < n; i += 256) s += sqerr[i];
  sm[threadIdx.x] = s;
  __syncthreads();
#pragma unroll
  for (int off = 128; off > 0; off >>= 1) {
    if (threadIdx.x < off) sm[threadIdx.x] += sm[threadIdx.x + off];
    __syncthreads();
  }
  if (threadIdx.x == 0) out[0] = sm[0] / (float)n;
}

extern "C" void kernel_launch(void* const* d_in, const int* in_sizes, int n_in,
                              void* d_out, int out_size, void* d_ws, size_t ws_size,
                              hipStream_t stream) {
  const float* ori = (const float*)d_in[0];
  const float* adv = (const float*)d_in[1];
  const float* nrm = (const float*)d_in[2];

  int B = in_sizes[0] / (3 * NPTS);      // 8 for the reference setup
  int total = B * NPTS;

  float* ori_kappa = (float*)d_ws;
  float* sqerr     = ori_kappa + total;

  dim3 grid(B * NTILES);
  ori_kappa_kernel<<<grid, 32, 0, stream>>>(ori, nrm, ori_kappa);
  adv_err_kernel<<<grid, 32, 0, stream>>>(ori, adv, nrm, ori_kappa, sqerr);
  reduce_kernel<<<1, 256, 0, stream>>>(sqerr, (float*)d_out, total);
}